// MyCorrWrapper_14551349199101
// MI455X (gfx1250) — compile-verified
//
#include <hip/hip_runtime.h>
#include <hip/hip_bf16.h>

typedef __attribute__((ext_vector_type(2))) float v2f;
typedef __attribute__((ext_vector_type(8))) float v8f;

#define RAD 4
#define PS  9   // patch size = 2*RAD+1

// One wave computes, for fixed (b, h, dh, 32-wide w span), the banded part of
// In1_h^T * In2_{h+dh-4} with V_WMMA_F32_16X16X4_F32 over K = C = 256.
//
// Register blocking: 2 M-tiles (w0, w0+16) x 3 N-tiles (w0-4, w0+12, w0+28):
//   acc0 = A0*B0 (dw = n-m)     acc1 = A0*B1 (dw = n-m+16)
//   acc2 = A1*B1 (dw = n-m)     acc3 = A1*B2 (dw = n-m+16)
// 10 loads : 4 WMMA per k-step; only 32 accumulator VGPRs so the scheduler
// has register headroom to keep many loads in flight (partial loadcnt waits).
// Zero padding handled branchlessly: clamped in-bounds addresses + 0/1 masks.
__global__ __launch_bounds__(32)
void corr_band_wmma_f32(const float* __restrict__ in1,
                        const float* __restrict__ in2,
                        float* __restrict__ out) {
    constexpr int B = 8, C = 256, H = 128, W = 256;
    constexpr int HW = H * W;

    const int lane = threadIdx.x;        // 0..31
    const int half = lane >> 4;          // 0 or 1
    const int i    = lane & 15;          // 0..15 (A: M index, B: N index)

    const int w0 = blockIdx.x << 5;      // 32-wide M span (2 tiles)
    const int h  = blockIdx.y;           // H
    const int dh = blockIdx.z % PS;      // 9 row displacements
    const int b  = blockIdx.z / PS;      // batch

    const int  h2  = h + dh - RAD;
    const bool vH  = (h2 >= 0) && (h2 < H);
    const int  h2c = vH ? h2 : 0;

    // ---- N-tile columns ----
    const int wq0 = w0 - 4 + i;          // N tile 0 (left clamp only)
    const int wq1 = w0 + 12 + i;         // N tile 1: always in [12, 251]
    const int wq2 = w0 + 28 + i;         // N tile 2 (right clamp only)
    const int c0  = max(wq0, 0);
    const int c2  = min(wq2, W - 1);

    const float m0 = (vH && wq0 >= 0) ? 1.0f : 0.0f;
    const float m1 = vH               ? 1.0f : 0.0f;
    const float m2 = (vH && wq2 < W)  ? 1.0f : 0.0f;

    // ---- A element offset: in1[b, c=k+2*half(+1), h, w0+i]; M1 via +16 imm
    uint32_t aoff = (uint32_t)((b * C + 2 * half) * H + h) * W + (uint32_t)(w0 + i);

    // ---- B element offsets: in2[b, c=k+2*half(+1), h2c, col] ----
    const uint32_t rowb = (uint32_t)((b * C + 2 * half) * H + h2c) * W;
    uint32_t b0off = rowb + (uint32_t)c0;
    uint32_t b1off = rowb + (uint32_t)wq1;
    uint32_t b2off = rowb + (uint32_t)c2;

    v8f acc0 = {0.f, 0.f, 0.f, 0.f, 0.f, 0.f, 0.f, 0.f};
    v8f acc1 = acc0, acc2 = acc0, acc3 = acc0;

    constexpr uint32_t kstep = 4u * HW;  // 4 channels per WMMA step

#pragma unroll 4
    for (int k = 0; k < C; k += 4) {
        v2f a0, a1, b0, b1, b2;
        a0.x = in1[aoff];
        a0.y = in1[aoff + HW];
        a1.x = in1[aoff + 16];
        a1.y = in1[aoff + 16 + HW];
        b0.x = in2[b0off]      * m0;
        b0.y = in2[b0off + HW] * m0;
        b1.x = in2[b1off]      * m1;
        b1.y = in2[b1off + HW] * m1;
        b2.x = in2[b2off]      * m2;
        b2.y = in2[b2off + HW] * m2;
        // (neg_a, A, neg_b, B, c_mod, C, reuse_a, reuse_b)
        acc0 = __builtin_amdgcn_wmma_f32_16x16x4_f32(false, a0, false, b0,
                                                     (short)0, acc0, false, false);
        acc1 = __builtin_amdgcn_wmma_f32_16x16x4_f32(false, a0, false, b1,
                                                     (short)0, acc1, false, false);
        acc2 = __builtin_amdgcn_wmma_f32_16x16x4_f32(false, a1, false, b1,
                                                     (short)0, acc2, false, false);
        acc3 = __builtin_amdgcn_wmma_f32_16x16x4_f32(false, a1, false, b2,
                                                     (short)0, acc3, false, false);
        aoff  += kstep;
        b0off += kstep;
        b1off += kstep;
        b2off += kstep;
    }

    // ---- Store the 9-diagonal band of each 16x16 tile ----
    // D layout: VGPR j holds M = j + 8*half, N = i.
    const float scale = 1.0f / (float)C;
    const uint32_t ob =
        (uint32_t)((b * (PS * PS) + dh * PS) * H + h) * W + (uint32_t)w0;

#pragma unroll
    for (int j = 0; j < 8; ++j) {
        const int m   = j + 8 * half;   // w offset inside each M tile
        const int dwA = i - m;          // acc0 (M0) and acc2 (M1)
        const int dwB = dwA + 16;       // acc1 (M0) and acc3 (M1)
        if (dwA >= 0 && dwA <= 2 * RAD) {
            out[ob + (uint32_t)dwA * HW + (uint32_t)m]        = acc0[j] * scale;
            out[ob + (uint32_t)dwA * HW + (uint32_t)(16 + m)] = acc2[j] * scale;
        }
        if (dwB >= 0 && dwB <= 2 * RAD) {
            out[ob + (uint32_t)dwB * HW + (uint32_t)m]        = acc1[j] * scale;
            out[ob + (uint32_t)dwB * HW + (uint32_t)(16 + m)] = acc3[j] * scale;
        }
    }
}

extern "C" void kernel_launch(void* const* d_in, const int* in_sizes, int n_in,
                              void* d_out, int out_size, void* d_ws, size_t ws_size,
                              hipStream_t stream) {
    (void)in_sizes; (void)n_in; (void)out_size; (void)d_ws; (void)ws_size;
    const float* in1 = (const float*)d_in[0];
    const float* in2 = (const float*)d_in[1];
    float*       out = (float*)d_out;

    dim3 grid(256 / 32, 128, 8 * PS);   // (8, 128, 72): one wave per (b,h,dh,32w)
    dim3 block(32);                     // one wave32
    hipLaunchKernelGGL(corr_band_wmma_f32, grid, block, 0, stream, in1, in2, out);
}